// VoxelPointsSampler_4123168604648
// MI455X (gfx1250) — compile-verified
//
#include <hip/hip_runtime.h>

typedef __attribute__((ext_vector_type(2))) float v2f;
typedef __attribute__((ext_vector_type(8))) float v8f;

#define GAMMA2 (1.1f * 1.1f)

// ---------------------------------------------------------------------------
// Kernel 0: zero a word region (used for presence bitmap and uniq array)
// ---------------------------------------------------------------------------
__global__ void vps_zero(unsigned* __restrict__ a, int n) {
    for (int i = blockIdx.x * blockDim.x + threadIdx.x; i < n;
         i += gridDim.x * blockDim.x)
        a[i] = 0u;
}

// ---------------------------------------------------------------------------
// Kernel 1: WMMA pairwise squared-distance mask.
// One wave owns one 32-point chunk (one output word) and loops over all box
// tiles. The full distance is computed inside the WMMA contraction:
//   A row m  = (-2*bx, -2*by, bq, 1)     (K = 0..3)
//   B col n  = ( px,    py,   1, pn)
//   => D[m][n] = bq + pn - 2*(bx*px + by*py) = dist^2
// Emits: B2[m][w] bit n -> (dist^2 <= r_m^2)
//        vmask[w] bit n -> any_m(dist^2 < r_m^2)
// No atomics: each wave fully owns its word column.
// ---------------------------------------------------------------------------
__global__ void vps_wmma_mask(const float* __restrict__ points,
                              const float* __restrict__ boxes,
                              unsigned* __restrict__ B2,
                              unsigned* __restrict__ vmaskArr,
                              int N, int M, int Mp, int W) {
    extern __shared__ float smem[];
    float* s_ap = smem;           // interleaved A pairs: 4*Mp floats
    float* s_r2 = smem + 4 * Mp;  // Mp floats (16B aligned: Mp % 16 == 0)
    for (int i = threadIdx.x; i < Mp; i += blockDim.x) {
        float bx = 0.f, by = 0.f, bq = 0.f, r2 = -1.f;
        if (i < M) {
            bx = boxes[i * 7 + 0];
            by = boxes[i * 7 + 1];
            float dx = boxes[i * 7 + 3];
            float dy = boxes[i * 7 + 4];
            bq = bx * bx + by * by;
            r2 = 0.25f * (dx * dx + dy * dy) * GAMMA2;
        }
        s_ap[2 * i + 0] = -2.0f * bx;        // K=0 coeff (lanes 0-15)
        s_ap[2 * i + 1] = -2.0f * by;        // K=1 coeff
        s_ap[2 * (Mp + i) + 0] = bq;         // K=2 coeff (lanes 16-31)
        s_ap[2 * (Mp + i) + 1] = 1.0f;       // K=3 coeff
        s_r2[i] = r2;
    }
    __syncthreads();

    int lane = threadIdx.x & 31;
    int wave = (blockIdx.x * blockDim.x + threadIdx.x) >> 5;
    if (wave >= W) return;  // wave-uniform exit
    int nb = wave * 32;

    int col = lane & 15;
    bool hi = lane >= 16;
    int n0 = nb + col;
    int n1 = nb + 16 + col;
    float px0 = 1e8f, py0 = 1e8f, px1 = 1e8f, py1 = 1e8f;
    if (n0 < N) { px0 = points[(size_t)n0 * 5 + 0]; py0 = points[(size_t)n0 * 5 + 1]; }
    if (n1 < N) { px1 = points[(size_t)n1 * 5 + 0]; py1 = points[(size_t)n1 * 5 + 1]; }
    float pn0 = px0 * px0 + py0 * py0;
    float pn1 = px1 * px1 + py1 * py1;
    // B fragment: lanes 0-15 carry K=0,1 = (px,py); lanes 16-31 K=2,3 = (1,pn)
    v2f b0, b1;
    b0.x = hi ? 1.0f : px0;
    b0.y = hi ? pn0 : py0;
    b1.x = hi ? 1.0f : px1;
    b1.y = hi ? pn1 : py1;

    bool lt0any = false, lt1any = false;
    int rbase = hi ? 8 : 0;  // this lane's row-group within the tile
    int numMt = Mp >> 4;
    for (int mt = 0; mt < numMt; ++mt) {
        int mb = mt * 16;
        // A fragment: single unconditional ds_load_b64
        int aidx = mb + col + (hi ? Mp : 0);
        v2f a = *(const v2f*)&s_ap[2 * aidx];
        v8f c0 = {};
        v8f c1 = {};
        c0 = __builtin_amdgcn_wmma_f32_16x16x4_f32(false, a, false, b0,
                                                   (short)0, c0, false, false);
        c1 = __builtin_amdgcn_wmma_f32_16x16x4_f32(false, a, false, b1,
                                                   (short)0, c1, false, false);
        // r^2 for this lane's 8 rows: two ds_load_b128
        float4 ra = *(const float4*)&s_r2[mb + rbase];
        float4 rb = *(const float4*)&s_r2[mb + rbase + 4];
        float rv[8] = {ra.x, ra.y, ra.z, ra.w, rb.x, rb.y, rb.z, rb.w};

        unsigned myWord = 0u;
#pragma unroll
        for (int r = 0; r < 8; ++r) {
            float d2_0 = c0[r];
            float d2_1 = c1[r];
            lt0any |= (d2_0 < rv[r]);
            lt1any |= (d2_1 < rv[r]);
            unsigned bal0 = __builtin_amdgcn_ballot_w32(d2_0 <= rv[r]);
            unsigned bal1 = __builtin_amdgcn_ballot_w32(d2_1 <= rv[r]);
            unsigned wLow = (bal0 & 0xFFFFu) | (bal1 << 16);
            unsigned wHigh = (bal0 >> 16) | (bal1 & 0xFFFF0000u);
            myWord = (lane == r) ? wLow : myWord;
            myWord = (lane == 8 + r) ? wHigh : myWord;
        }
        int mrow = mb + lane;
        if (lane < 16 && mrow < M)
            B2[(size_t)mrow * W + wave] = myWord;  // single store per mtile
    }
    unsigned balv0 = __builtin_amdgcn_ballot_w32(lt0any);
    unsigned balv1 = __builtin_amdgcn_ballot_w32(lt1any);
    unsigned vword = ((balv0 & 0xFFFFu) | (balv0 >> 16)) |
                     (((balv1 & 0xFFFFu) | (balv1 >> 16)) << 16);
    if (lane == 0) vmaskArr[wave] = vword;
}

// ---------------------------------------------------------------------------
// Kernel 2: per-box ordered selection (top_k over 0/1 mask semantics):
// first S set-bit indices ascending; remaining slots filled with the first
// clear-bit indices ascending. One 32-thread wave per box.
// ---------------------------------------------------------------------------
__global__ void vps_select(const unsigned* __restrict__ B2,
                           const unsigned* __restrict__ vmaskArr,
                           int* __restrict__ sidx, int* __restrict__ cnt,
                           int N, int S, int W) {
    extern __shared__ int sh[];
    int* scan1 = sh;        // 32
    int* scan0 = sh + 32;   // 32
    int* zlist = sh + 64;   // S
    int m = blockIdx.x;
    int t = threadIdx.x;
    int total1 = 0, total0 = 0;
    for (int base = 0; base < W; base += 32) {
        int wi = base + t;
        unsigned word = 0u;
        int valid = 0;
        if (wi < W) {
            word = B2[(size_t)m * W + wi] & vmaskArr[wi];
            valid = (wi == W - 1) ? (N - (W - 1) * 32) : 32;
        }
        unsigned vbits = (valid >= 32) ? 0xFFFFFFFFu : ((1u << valid) - 1u);
        word &= vbits;
        int c1 = __popc(word);
        int c0 = valid - c1;
        scan1[t] = c1;
        scan0[t] = c0;
        __syncthreads();
        for (int d = 1; d < 32; d <<= 1) {
            int a1 = (t >= d) ? scan1[t - d] : 0;
            int a0 = (t >= d) ? scan0[t - d] : 0;
            __syncthreads();
            scan1[t] += a1;
            scan0[t] += a0;
            __syncthreads();
        }
        int pre1 = total1 + scan1[t] - c1;  // exclusive slot base (ones)
        int pre0 = total0 + scan0[t] - c0;  // exclusive slot base (zeros)
        int wsum1 = scan1[31];
        int wsum0 = scan0[31];
        if (pre1 < S) {
            unsigned x = word;
            int k = 0;
            while (x) {
                int p = __ffs(x) - 1;
                x &= x - 1;
                int slot = pre1 + k;
                if (slot < S) sidx[(size_t)m * S + slot] = wi * 32 + p;
                ++k;
            }
        }
        if (pre0 < S) {
            unsigned x = (~word) & vbits;
            int k = 0;
            while (x) {
                int p = __ffs(x) - 1;
                x &= x - 1;
                int slot = pre0 + k;
                if (slot < S) zlist[slot] = wi * 32 + p;
                ++k;
            }
        }
        total1 += wsum1;
        total0 += wsum0;
        __syncthreads();
    }
    int c = total1 < S ? total1 : S;
    if (t == 0) cnt[m] = c;
    __syncthreads();
    for (int s = c + t; s < S; s += 32)
        sidx[(size_t)m * S + s] = zlist[s - c];
}

// ---------------------------------------------------------------------------
// Kernel 3: mark presence bitmap for unique()
// ---------------------------------------------------------------------------
__global__ void vps_mark(const int* __restrict__ sidx,
                         unsigned* __restrict__ bitmap, int K) {
    int j = blockIdx.x * blockDim.x + threadIdx.x;
    if (j >= K) return;
    int v = sidx[j];
    atomicOr(&bitmap[v >> 5], 1u << (v & 31));
}

// ---------------------------------------------------------------------------
// Kernel 4: exclusive prefix of per-word popcounts (single block)
// ---------------------------------------------------------------------------
__global__ void vps_scan(const unsigned* __restrict__ bitmap,
                         int* __restrict__ wordPrefix,
                         int* __restrict__ uCount, int W) {
    __shared__ int sums[1024];
    int t = threadIdx.x;
    int T = blockDim.x;
    int chunk = (W + T - 1) / T;
    int lo = t * chunk;
    int hiE = lo + chunk;
    if (lo > W) lo = W;
    if (hiE > W) hiE = W;
    int s = 0;
    for (int i = lo; i < hiE; ++i) s += __popc(bitmap[i]);
    sums[t] = s;
    __syncthreads();
    for (int d = 1; d < T; d <<= 1) {
        int v = (t >= d) ? sums[t - d] : 0;
        __syncthreads();
        sums[t] += v;
        __syncthreads();
    }
    int run = sums[t] - s;  // exclusive base
    for (int i = lo; i < hiE; ++i) {
        wordPrefix[i] = run;
        run += __popc(bitmap[i]);
    }
    if (t == T - 1) uCount[0] = sums[T - 1];
}

// ---------------------------------------------------------------------------
// Kernel 5: expand bitmap into sorted unique index array (select)
// ---------------------------------------------------------------------------
__global__ void vps_expand(const unsigned* __restrict__ bitmap,
                           const int* __restrict__ wordPrefix,
                           int* __restrict__ uniqArr, int W, int cap) {
    int i = blockIdx.x * blockDim.x + threadIdx.x;
    if (i >= W) return;
    unsigned x = bitmap[i];
    int pos = wordPrefix[i];
    while (x) {
        int p = __ffs(x) - 1;
        x &= x - 1;
        if (pos < cap) uniqArr[pos] = i * 32 + p;
        ++pos;
    }
}

// ---------------------------------------------------------------------------
// Kernel 6: sampled_points (masked gather) + idx (inverse rank via bitmap)
// ---------------------------------------------------------------------------
__global__ void vps_gather_sampled(const float* __restrict__ points,
                                   const int* __restrict__ sidx,
                                   const int* __restrict__ cnt,
                                   const unsigned* __restrict__ bitmap,
                                   const int* __restrict__ wordPrefix,
                                   float* __restrict__ outPts,
                                   float* __restrict__ outIdx, int M, int S) {
    int j = blockIdx.x * blockDim.x + threadIdx.x;
    if (j >= M * S) return;
    int m = j / S;
    int s = j - m * S;
    bool valid = s < cnt[m];
    float o0 = 0.f, o1 = 0.f, o2 = 0.f, o3 = 0.f, o4 = 0.f, iv = 0.f;
    if (valid) {
        int v = sidx[j];
        o0 = points[(size_t)v * 5 + 0];
        o1 = points[(size_t)v * 5 + 1];
        o2 = points[(size_t)v * 5 + 2];
        o3 = points[(size_t)v * 5 + 3];
        o4 = points[(size_t)v * 5 + 4];
        unsigned low = (1u << (v & 31)) - 1u;
        int rank = wordPrefix[v >> 5] + __popc(bitmap[v >> 5] & low);
        iv = (float)rank;
    }
    outPts[(size_t)j * 5 + 0] = o0;
    outPts[(size_t)j * 5 + 1] = o1;
    outPts[(size_t)j * 5 + 2] = o2;
    outPts[(size_t)j * 5 + 3] = o3;
    outPts[(size_t)j * 5 + 4] = o4;
    outIdx[j] = iv;
}

// ---------------------------------------------------------------------------
// Kernel 7: query_points gather (padded slots -> uniqArr==0 -> points[0])
// ---------------------------------------------------------------------------
__global__ void vps_gather_query(const float* __restrict__ points,
                                 const int* __restrict__ uniqArr,
                                 float* __restrict__ outQ, int K) {
    int j = blockIdx.x * blockDim.x + threadIdx.x;
    if (j >= K) return;
    int u = uniqArr[j];
#pragma unroll
    for (int c = 0; c < 5; ++c)
        outQ[(size_t)j * 5 + c] = points[(size_t)u * 5 + c];
}

extern "C" void kernel_launch(void* const* d_in, const int* in_sizes, int n_in,
                              void* d_out, int out_size, void* d_ws,
                              size_t ws_size, hipStream_t stream) {
    const float* points = (const float*)d_in[0];
    const float* boxes = (const float*)d_in[1];
    // d_in[2] = pc_start (cancels out of the distance math), d_in[3] = S (derived)

    const int N = in_sizes[0] / 5;
    const int M = in_sizes[1] / 7;
    const int S = out_size / (M * 11);  // out = M*S*5 + M*S + M*S*5
    const int W = (N + 31) / 32;
    const int Mp = ((M + 15) / 16) * 16;
    const int MS = M * S;

    // workspace carve (all 4-byte types)
    char* p = (char*)d_ws;
    unsigned* B2 = (unsigned*)p;        p += (size_t)M * W * 4;
    unsigned* vmaskA = (unsigned*)p;    p += (size_t)W * 4;
    unsigned* bitmap = (unsigned*)p;    p += (size_t)W * 4;
    int* wordPrefix = (int*)p;          p += (size_t)W * 4;
    int* sidx = (int*)p;                p += (size_t)MS * 4;
    int* cnt = (int*)p;                 p += (size_t)M * 4;
    int* uniqArr = (int*)p;             p += (size_t)MS * 4;
    int* uCount = (int*)p;              p += 4;

    float* outPts = (float*)d_out;
    float* outIdx = outPts + (size_t)MS * 5;
    float* outQ = outIdx + (size_t)MS;

    // 0: zero bitmap + uniq array (re-done every call -> deterministic)
    vps_zero<<<(W + 255) / 256, 256, 0, stream>>>(bitmap, W);
    vps_zero<<<(MS + 255) / 256, 256, 0, stream>>>((unsigned*)uniqArr, MS);

    // 1: WMMA pairwise mask
    {
        int threads = W * 32;
        int block = 128;
        int grid = (threads + block - 1) / block;
        size_t shmem = (size_t)5 * Mp * sizeof(float);
        vps_wmma_mask<<<grid, block, shmem, stream>>>(points, boxes, B2,
                                                      vmaskA, N, M, Mp, W);
    }

    // 2: per-box ordered selection
    {
        size_t shmem = (size_t)(64 + S) * sizeof(int);
        vps_select<<<M, 32, shmem, stream>>>(B2, vmaskA, sidx, cnt, N, S, W);
    }

    // 3: mark presence bitmap
    vps_mark<<<(MS + 255) / 256, 256, 0, stream>>>(sidx, bitmap, MS);

    // 4: popcount prefix scan
    vps_scan<<<1, 1024, 0, stream>>>(bitmap, wordPrefix, uCount, W);

    // 5: expand sorted unique
    vps_expand<<<(W + 255) / 256, 256, 0, stream>>>(bitmap, wordPrefix,
                                                    uniqArr, W, MS);

    // 6: sampled_points + idx
    vps_gather_sampled<<<(MS + 127) / 128, 128, 0, stream>>>(
        points, sidx, cnt, bitmap, wordPrefix, outPts, outIdx, M, S);

    // 7: query_points
    vps_gather_query<<<(MS + 127) / 128, 128, 0, stream>>>(points, uniqArr,
                                                           outQ, MS);
}